// TransformerBlock_70179765616653
// MI455X (gfx1250) — compile-verified
//
#include <hip/hip_runtime.h>
#include <math.h>

#define BATCH 8
#define CH 64
#define CH2 128
#define HH 224
#define WW 224
#define HW 50176            // 224*224
#define NPIX (BATCH * HW)   // 401408

typedef __attribute__((ext_vector_type(2))) float v2f;
typedef __attribute__((ext_vector_type(8))) float v8f;

__device__ __forceinline__ v8f wmma4(v2f a, v2f b, v8f c) {
  // V_WMMA_F32_16X16X4_F32 : D(16x16,f32) = A(16x4,f32) * B(4x16,f32) + C
  return __builtin_amdgcn_wmma_f32_16x16x4_f32(false, a, false, b, (short)0, c,
                                               false, false);
}

__device__ __constant__ int MXc[16] = {0,0,6,0,0,1,1,4,5,1,3,0,0,0,3,2};
__device__ __constant__ int MYc[16] = {0,1,0,5,2,0,2,0,0,6,0,4,6,3,2,5};

// ---------------------------------------------------------------- utilities
__device__ __forceinline__ float tap9(const float* __restrict__ p,
                                      const float* __restrict__ k,
                                      int r, int col) {
  float acc = 0.f;
#pragma unroll
  for (int dy = -1; dy <= 1; ++dy) {
    int rr = r + dy;
    if (rr < 0 || rr >= HH) continue;
#pragma unroll
    for (int dx = -1; dx <= 1; ++dx) {
      int cc = col + dx;
      if (cc < 0 || cc >= WW) continue;
      acc += k[(dy + 1) * 3 + (dx + 1)] * p[rr * WW + cc];
    }
  }
  return acc;
}

// ------------------------------------------------------------------ kernels
__global__ __launch_bounds__(256) void k_zero(float* stats) {
  for (int i = threadIdx.x; i < 1536; i += 256) stats[i] = 0.f;
}

// channel-LayerNorm + 1x1 conv (64 -> 128) via f32 WMMA.
// HAS_BIAS is compile-time so the store epilogue has no branches; bias lives
// in LDS so there are no global loads between the WMMA chain and the stores.
template <bool HAS_BIAS>
__global__ __launch_bounds__(256) void k_ln_conv(
    const float* __restrict__ src, const float* __restrict__ lnw,
    const float* __restrict__ lnb, float eps, const float* __restrict__ wmat,
    const float* __restrict__ bias, float* __restrict__ dst) {
  __shared__ float wl[128 * 68];
  __shared__ float lw[64], lb[64], bb[128];
  int tid = threadIdx.x;
  for (int i = tid; i < 128 * 64; i += 256) wl[(i >> 6) * 68 + (i & 63)] = wmat[i];
  if (tid < 64) { lw[tid] = lnw[tid]; lb[tid] = lnb[tid]; }
  if (HAS_BIAS) {
    if (tid < 128) bb[tid] = bias[tid];
  }
  __syncthreads();

  int wave = tid >> 5, lane = tid & 31;
  int m = lane & 15, kh = (lane >> 4) << 1, oh = (lane >> 4) << 3;
  long long p0 = (long long)blockIdx.x * 128 + wave * 16;
  int b = (int)(p0 / HW);
  int pix = (int)(p0 % HW) + m;
  const float* sb = src + (long long)b * 64 * HW;

  v2f bf[16];
  float s1 = 0.f, s2 = 0.f;
#pragma unroll
  for (int s = 0; s < 16; ++s) {
    int c0 = 4 * s + kh;
    float a0 = sb[(long long)c0 * HW + pix];
    float a1 = sb[(long long)(c0 + 1) * HW + pix];
    bf[s].x = a0; bf[s].y = a1;
    s1 += a0 + a1; s2 += a0 * a0 + a1 * a1;
  }
  s1 += __shfl_xor(s1, 16, 32);
  s2 += __shfl_xor(s2, 16, 32);
  float mu = s1 * (1.f / 64.f);
  float var = s2 * (1.f / 64.f) - mu * mu;
  float rstd = rsqrtf(var + eps);
#pragma unroll
  for (int s = 0; s < 16; ++s) {
    int c0 = 4 * s + kh;
    bf[s].x = (bf[s].x - mu) * rstd * lw[c0] + lb[c0];
    bf[s].y = (bf[s].y - mu) * rstd * lw[c0 + 1] + lb[c0 + 1];
  }
  float* db = dst + (long long)b * 128 * HW;
#pragma unroll
  for (int n = 0; n < 8; ++n) {
    v8f acc = {};
#pragma unroll
    for (int s = 0; s < 16; ++s) {
      int c0 = 4 * s + kh;
      v2f a;
      a.x = wl[(n * 16 + m) * 68 + c0];
      a.y = wl[(n * 16 + m) * 68 + c0 + 1];
      acc = wmma4(a, bf[s], acc);
    }
#pragma unroll
    for (int j = 0; j < 8; ++j) {
      int oc = n * 16 + j + oh;
      float v = acc[j];
      if (HAS_BIAS) v += bb[oc];
      db[(long long)oc * HW + pix] = v;
    }
  }
}

// depthwise 3x3 (+bias) over 128 channels, accumulating instnorm stats (c<64)
__global__ __launch_bounds__(256) void k_dwconv_stats(
    const float* __restrict__ src, const float* __restrict__ w9,
    const float* __restrict__ bias, float* __restrict__ dst,
    float* __restrict__ sumA, float* __restrict__ sumQ) {
  __shared__ float r1[256], r2[256];
  int blk = blockIdx.x;                 // B*128*8
  int stripe = blk & 7;
  int c = (blk >> 3) & 127;
  int b = blk >> 10;
  const float* sp = src + ((long long)b * 128 + c) * HW;
  float* dp = dst + ((long long)b * 128 + c) * HW;
  float k[9];
#pragma unroll
  for (int i = 0; i < 9; ++i) k[i] = w9[c * 9 + i];
  float bc = bias[c];
  float ls = 0.f, lq = 0.f;
  for (int i = threadIdx.x; i < 28 * WW; i += 256) {
    int r = stripe * 28 + i / WW;
    int col = i % WW;
    float v = bc + tap9(sp, k, r, col);
    dp[r * WW + col] = v;
    ls += v; lq += v * v;
  }
  if (c < 64) {
    int tid = threadIdx.x;
    r1[tid] = ls; r2[tid] = lq;
    __syncthreads();
    for (int s = 128; s > 0; s >>= 1) {
      if (tid < s) { r1[tid] += r1[tid + s]; r2[tid] += r2[tid + s]; }
      __syncthreads();
    }
    if (tid == 0) {
      atomicAdd(&sumA[b * 64 + c], r1[0]);
      atomicAdd(&sumQ[b * 64 + c], r2[0]);
    }
  }
}

// t = instnorm(a)*g, plus DCT channel-attention reduction
__global__ __launch_bounds__(256) void k_gate_dct(
    const float* __restrict__ u, const float* __restrict__ sumA,
    const float* __restrict__ sumQ, const float* __restrict__ inw,
    const float* __restrict__ inb, float* __restrict__ t,
    float* __restrict__ ydct) {
  __shared__ float r1[256];
  int blk = blockIdx.x;                 // B*64*8
  int stripe = blk & 7;
  int c = (blk >> 3) & 63;
  int b = blk >> 9;
  float mean = sumA[b * 64 + c] * (1.f / (float)HW);
  float var = sumQ[b * 64 + c] * (1.f / (float)HW) - mean * mean;
  float rstd = rsqrtf(var + 1e-5f);
  float sw = inw[c], sb = inb[c];
  const float* ap = u + ((long long)b * 128 + c) * HW;
  const float* gp = u + ((long long)b * 128 + c + 64) * HW;
  float* tp = t + ((long long)b * 64 + c) * HW;
  int fi = c >> 2;
  float ux = (float)(MXc[fi] * 8), uy = (float)(MYc[fi] * 8);
  const float rs56 = 0.13363062f, SQ2 = 1.41421356f, PI = 3.14159265358979f;
  float fh = rs56 * (ux != 0.f ? SQ2 : 1.f);
  float fw = rs56 * (uy != 0.f ? SQ2 : 1.f);
  float sc = fh * fw * (1.f / 16.f);
  float cph = PI * ux / 56.f, cpw = PI * uy / 56.f;
  float lacc = 0.f;
  for (int i = threadIdx.x; i < 28 * WW; i += 256) {
    int r = stripe * 28 + i / WW;
    int col = i % WW;
    float a = ap[r * WW + col], g = gp[r * WW + col];
    float tv = ((a - mean) * rstd * sw + sb) * g;
    tp[r * WW + col] = tv;
    float h4 = (float)(r >> 2) + 0.5f, w4 = (float)(col >> 2) + 0.5f;
    lacc += tv * __cosf(cph * h4) * __cosf(cpw * w4) * sc;
  }
  int tid = threadIdx.x;
  r1[tid] = lacc;
  __syncthreads();
  for (int s = 128; s > 0; s >>= 1) {
    if (tid < s) r1[tid] += r1[tid + s];
    __syncthreads();
  }
  if (tid == 0) atomicAdd(&ydct[b * 64 + c], r1[0]);
}

// SE MLP: z = sigmoid(relu(y @ fc1^T) @ fc2^T)
__global__ __launch_bounds__(256) void k_se(const float* __restrict__ ydct,
                                            const float* __restrict__ fc1,
                                            const float* __restrict__ fc2,
                                            float* __restrict__ z) {
  __shared__ float mid[8][4];
  int tid = threadIdx.x;
  if (tid < 32) {
    int b = tid >> 2, m = tid & 3;
    float a = 0.f;
    for (int c = 0; c < 64; ++c) a += ydct[b * 64 + c] * fc1[m * 64 + c];
    mid[b][m] = fmaxf(a, 0.f);
  }
  __syncthreads();
  for (int idx = tid; idx < 512; idx += 256) {
    int b = idx >> 6, c = idx & 63;
    float a = 0.f;
#pragma unroll
    for (int m = 0; m < 4; ++m) a += mid[b][m] * fc2[c * 4 + m];
    z[idx] = 1.f / (1.f + __expf(-a));
  }
}

// fused: t*z -> conv3 -> residual -> LN -> conv4 -> gate -> conv5 -> y
__global__ __launch_bounds__(256) void k_mid(
    const float* __restrict__ t, const float* __restrict__ zs_,
    const float* __restrict__ x, const float* __restrict__ w3,
    const float* __restrict__ b3_, const float* __restrict__ beta_,
    const float* __restrict__ nw_, const float* __restrict__ nb_,
    const float* __restrict__ w4, const float* __restrict__ b4_,
    const float* __restrict__ w5, const float* __restrict__ b5_,
    float* __restrict__ y) {
  __shared__ float TT[128 * 68];
  __shared__ float zs[64], b3[64], bet[64], nw[64], nb[64], b5[64], b4[128];
  int tid = threadIdx.x;
  long long base = (long long)blockIdx.x * 128;
  int b = (int)(base / HW);
  int pbase = (int)(base % HW);
  if (tid < 64) {
    zs[tid] = zs_[b * 64 + tid]; b3[tid] = b3_[tid]; bet[tid] = beta_[tid];
    nw[tid] = nw_[tid]; nb[tid] = nb_[tid]; b5[tid] = b5_[tid];
  }
  if (tid < 128) b4[tid] = b4_[tid];
  __syncthreads();
  const float* tbp = t + (long long)b * 64 * HW + pbase;
  for (int i = tid; i < 128 * 64; i += 256) {
    int ch = i >> 7, px = i & 127;
    TT[px * 68 + ch] = tbp[(long long)ch * HW + px] * zs[ch];
  }
  __syncthreads();

  int wave = tid >> 5, lane = tid & 31;
  int m = lane & 15, kh = (lane >> 4) << 1, oh = (lane >> 4) << 3;
  float* myT = TT + wave * 16 * 68;
  int pix = pbase + wave * 16 + m;

  // conv3 (64->64)
  float v[4][8];
#pragma unroll
  for (int n = 0; n < 4; ++n) {
    v8f acc = {};
#pragma unroll
    for (int s = 0; s < 16; ++s) {
      int c0 = 4 * s + kh;
      v2f a, bfr;
      a.x = w3[(n * 16 + m) * 64 + c0]; a.y = w3[(n * 16 + m) * 64 + c0 + 1];
      bfr.x = myT[m * 68 + c0]; bfr.y = myT[m * 68 + c0 + 1];
      acc = wmma4(a, bfr, acc);
    }
#pragma unroll
    for (int j = 0; j < 8; ++j) v[n][j] = acc[j];
  }
  // residual + channel LN (eps 1e-6)
  const float* xb = x + (long long)b * 64 * HW;
  float rr[4][8];
  float s1 = 0.f, s2 = 0.f;
#pragma unroll
  for (int n = 0; n < 4; ++n)
#pragma unroll
    for (int j = 0; j < 8; ++j) {
      int oc = n * 16 + j + oh;
      float rv = xb[(long long)oc * HW + pix] + (v[n][j] + b3[oc]) * bet[oc];
      rr[n][j] = rv;
      s1 += rv; s2 += rv * rv;
    }
  s1 += __shfl_xor(s1, 16, 32);
  s2 += __shfl_xor(s2, 16, 32);
  float mu = s1 * (1.f / 64.f);
  float var = s2 * (1.f / 64.f) - mu * mu;
  float rstd = rsqrtf(var + 1e-6f);
  __syncthreads();
#pragma unroll
  for (int n = 0; n < 4; ++n)
#pragma unroll
    for (int j = 0; j < 8; ++j) {
      int oc = n * 16 + j + oh;
      myT[m * 68 + oc] = (rr[n][j] - mu) * rstd * nw[oc] + nb[oc];
    }
  __syncthreads();
  // conv4 (64->128) + SimpleGate, results kept in regs then written back
  float garr[4][8];
#pragma unroll
  for (int n = 0; n < 4; ++n) {
    v8f aA = {}, aB = {};
#pragma unroll
    for (int s = 0; s < 16; ++s) {
      int c0 = 4 * s + kh;
      v2f bfr; bfr.x = myT[m * 68 + c0]; bfr.y = myT[m * 68 + c0 + 1];
      v2f a1; a1.x = w4[(n * 16 + m) * 64 + c0];
      a1.y = w4[(n * 16 + m) * 64 + c0 + 1];
      aA = wmma4(a1, bfr, aA);
      v2f a2; a2.x = w4[((n + 4) * 16 + m) * 64 + c0];
      a2.y = w4[((n + 4) * 16 + m) * 64 + c0 + 1];
      aB = wmma4(a2, bfr, aB);
    }
#pragma unroll
    for (int j = 0; j < 8; ++j) {
      int c = n * 16 + j + oh;
      garr[n][j] = (aA[j] + b4[c]) * (aB[j] + b4[c + 64]);
    }
  }
  __syncthreads();
#pragma unroll
  for (int n = 0; n < 4; ++n)
#pragma unroll
    for (int j = 0; j < 8; ++j) myT[m * 68 + n * 16 + j + oh] = garr[n][j];
  __syncthreads();
  // conv5 (64->64)
  float* yb = y + (long long)b * 64 * HW;
#pragma unroll
  for (int n = 0; n < 4; ++n) {
    v8f acc = {};
#pragma unroll
    for (int s = 0; s < 16; ++s) {
      int c0 = 4 * s + kh;
      v2f a, bfr;
      a.x = w5[(n * 16 + m) * 64 + c0]; a.y = w5[(n * 16 + m) * 64 + c0 + 1];
      bfr.x = myT[m * 68 + c0]; bfr.y = myT[m * 68 + c0 + 1];
      acc = wmma4(a, bfr, acc);
    }
#pragma unroll
    for (int j = 0; j < 8; ++j) {
      int oc = n * 16 + j + oh;
      yb[(long long)oc * HW + pix] = acc[j] + b5[oc];
    }
  }
}

// per-8x8-patch rfft2 * fft_w -> irfft2 (one thread per patch)
__global__ __launch_bounds__(256) void k_pfft(const float* __restrict__ h,
                                              const float* __restrict__ fw,
                                              float* __restrict__ hf) {
  const float C8[8] = {1.f, 0.70710678f, 0.f, -0.70710678f,
                       -1.f, -0.70710678f, 0.f, 0.70710678f};
  const float S8[8] = {0.f, 0.70710678f, 1.f, 0.70710678f,
                       0.f, -0.70710678f, -1.f, -0.70710678f};
  long long idx = (long long)blockIdx.x * 256 + threadIdx.x;  // B*128*28*28
  int px = (int)(idx % 28); idx /= 28;
  int py = (int)(idx % 28); idx /= 28;
  int c = (int)(idx % 128);
  int b = (int)(idx / 128);
  long long off = ((long long)b * 128 + c) * HW + (py * 8) * WW + px * 8;
  const float* sp = h + off;
  float xr[8][8];
#pragma unroll
  for (int r = 0; r < 8; ++r) {
    float4 lo = *(const float4*)(sp + r * WW);
    float4 hi = *(const float4*)(sp + r * WW + 4);
    xr[r][0] = lo.x; xr[r][1] = lo.y; xr[r][2] = lo.z; xr[r][3] = lo.w;
    xr[r][4] = hi.x; xr[r][5] = hi.y; xr[r][6] = hi.z; xr[r][7] = hi.w;
  }
  float Fr[8][5], Fi[8][5];
#pragma unroll
  for (int r = 0; r < 8; ++r)
#pragma unroll
    for (int kx = 0; kx < 5; ++kx) {
      float fr = 0.f, fi = 0.f;
#pragma unroll
      for (int w = 0; w < 8; ++w) {
        int ph = (w * kx) & 7;
        fr += xr[r][w] * C8[ph];
        fi -= xr[r][w] * S8[ph];
      }
      Fr[r][kx] = fr; Fi[r][kx] = fi;
    }
  float Gr[8][5], Gi[8][5];
#pragma unroll
  for (int ky = 0; ky < 8; ++ky)
#pragma unroll
    for (int kx = 0; kx < 5; ++kx) {
      float gr = 0.f, gi = 0.f;
#pragma unroll
      for (int r = 0; r < 8; ++r) {
        int ph = (r * ky) & 7;
        gr += Fr[r][kx] * C8[ph] + Fi[r][kx] * S8[ph];
        gi += Fi[r][kx] * C8[ph] - Fr[r][kx] * S8[ph];
      }
      float wv = fw[((long long)c * 8 + ky) * 5 + kx];
      Gr[ky][kx] = gr * wv; Gi[ky][kx] = gi * wv;
    }
  float Hr[8][5], Hi[8][5];
#pragma unroll
  for (int r = 0; r < 8; ++r)
#pragma unroll
    for (int kx = 0; kx < 5; ++kx) {
      float hr = 0.f, hi = 0.f;
#pragma unroll
      for (int ky = 0; ky < 8; ++ky) {
        int ph = (ky * r) & 7;
        hr += Gr[ky][kx] * C8[ph] - Gi[ky][kx] * S8[ph];
        hi += Gr[ky][kx] * S8[ph] + Gi[ky][kx] * C8[ph];
      }
      Hr[r][kx] = hr * 0.125f; Hi[r][kx] = hi * 0.125f;
    }
  float* dp = hf + off;
#pragma unroll
  for (int r = 0; r < 8; ++r) {
    float o[8];
#pragma unroll
    for (int w = 0; w < 8; ++w) {
      float acc = Hr[r][0] + Hr[r][4] * C8[(4 * w) & 7];
#pragma unroll
      for (int k = 1; k < 4; ++k) {
        int ph = (k * w) & 7;
        acc += 2.f * (Hr[r][k] * C8[ph] - Hi[r][k] * S8[ph]);
      }
      o[w] = acc * 0.125f;
    }
    *(float4*)(dp + r * WW) = make_float4(o[0], o[1], o[2], o[3]);
    *(float4*)(dp + r * WW + 4) = make_float4(o[4], o[5], o[6], o[7]);
  }
}

// depthwise 3x3 + exact-GELU SimpleGate
__global__ __launch_bounds__(256) void k_dwgate(const float* __restrict__ hf,
                                                const float* __restrict__ w9,
                                                float* __restrict__ gv) {
  int blk = blockIdx.x;                 // B*64*8
  int stripe = blk & 7;
  int c = (blk >> 3) & 63;
  int b = blk >> 9;
  const float* p1 = hf + ((long long)b * 128 + c) * HW;
  const float* p2 = hf + ((long long)b * 128 + c + 64) * HW;
  float k1[9], k2[9];
#pragma unroll
  for (int i = 0; i < 9; ++i) { k1[i] = w9[c * 9 + i]; k2[i] = w9[(c + 64) * 9 + i]; }
  float* dp = gv + ((long long)b * 64 + c) * HW;
  for (int i = threadIdx.x; i < 28 * WW; i += 256) {
    int r = stripe * 28 + i / WW;
    int col = i % WW;
    float d1 = tap9(p1, k1, r, col);
    float d2 = tap9(p2, k2, r, col);
    float ge = 0.5f * d1 * (1.f + erff(d1 * 0.70710678f));
    dp[r * WW + col] = ge * d2;
  }
}

// pout (64->64, WMMA) + y -> out
__global__ __launch_bounds__(256) void k_conv_add(
    const float* __restrict__ gv, const float* __restrict__ wmat,
    const float* __restrict__ addy, float* __restrict__ dst) {
  __shared__ float wl[64 * 68];
  int tid = threadIdx.x;
  for (int i = tid; i < 64 * 64; i += 256) wl[(i >> 6) * 68 + (i & 63)] = wmat[i];
  __syncthreads();
  int wave = tid >> 5, lane = tid & 31;
  int m = lane & 15, kh = (lane >> 4) << 1, oh = (lane >> 4) << 3;
  long long p0 = (long long)blockIdx.x * 128 + wave * 16;
  int b = (int)(p0 / HW);
  int pix = (int)(p0 % HW) + m;
  const float* gb = gv + (long long)b * 64 * HW;
  v2f bf[16];
#pragma unroll
  for (int s = 0; s < 16; ++s) {
    int c0 = 4 * s + kh;
    bf[s].x = gb[(long long)c0 * HW + pix];
    bf[s].y = gb[(long long)(c0 + 1) * HW + pix];
  }
  const float* ab = addy + (long long)b * 64 * HW;
  float* db = dst + (long long)b * 64 * HW;
#pragma unroll
  for (int n = 0; n < 4; ++n) {
    v8f acc = {};
#pragma unroll
    for (int s = 0; s < 16; ++s) {
      int c0 = 4 * s + kh;
      v2f a;
      a.x = wl[(n * 16 + m) * 68 + c0];
      a.y = wl[(n * 16 + m) * 68 + c0 + 1];
      acc = wmma4(a, bf[s], acc);
    }
#pragma unroll
    for (int j = 0; j < 8; ++j) {
      int oc = n * 16 + j + oh;
      db[(long long)oc * HW + pix] = acc[j] + ab[(long long)oc * HW + pix];
    }
  }
}

// ------------------------------------------------------------------- launch
extern "C" void kernel_launch(void* const* d_in, const int* in_sizes, int n_in,
                              void* d_out, int out_size, void* d_ws,
                              size_t ws_size, hipStream_t stream) {
  const float* x       = (const float*)d_in[0];
  const float* n1_w    = (const float*)d_in[1];
  const float* n1_b    = (const float*)d_in[2];
  const float* conv1_w = (const float*)d_in[3];
  const float* conv1_b = (const float*)d_in[4];
  const float* conv2_w = (const float*)d_in[5];
  const float* conv2_b = (const float*)d_in[6];
  const float* in_w    = (const float*)d_in[7];
  const float* in_b    = (const float*)d_in[8];
  const float* fc1_w   = (const float*)d_in[9];
  const float* fc2_w   = (const float*)d_in[10];
  const float* conv3_w = (const float*)d_in[11];
  const float* conv3_b = (const float*)d_in[12];
  const float* beta    = (const float*)d_in[13];
  const float* n2n_w   = (const float*)d_in[14];
  const float* n2n_b   = (const float*)d_in[15];
  const float* conv4_w = (const float*)d_in[16];
  const float* conv4_b = (const float*)d_in[17];
  const float* conv5_w = (const float*)d_in[18];
  const float* conv5_b = (const float*)d_in[19];
  const float* ln_w    = (const float*)d_in[20];
  const float* ln_b    = (const float*)d_in[21];
  const float* pin_w   = (const float*)d_in[22];
  const float* dw_w    = (const float*)d_in[23];
  const float* fft_w   = (const float*)d_in[24];
  const float* pout_w  = (const float*)d_in[25];
  float* out = (float*)d_out;

  float* ws = (float*)d_ws;
  const size_t F1 = (size_t)BATCH * 128 * HW;   // 51,380,224 floats
  const size_t F2 = (size_t)BATCH * 64 * HW;    // 25,690,112 floats
  float* w1 = ws;                  // [B,128,HW]  (reused as h)
  float* u  = ws + F1;             // [B,128,HW]  (reused as hf)
  float* tb = ws + 2 * F1;         // [B,64,HW]   (reused as gv)
  float* yb = ws + 2 * F1 + F2;    // [B,64,HW]
  float* stats = yb + F2;          // sumA(512) sumQ(512) ydct(512) z(512)
  float* sumA = stats;
  float* sumQ = stats + 512;
  float* ydct = stats + 1024;
  float* zsc  = stats + 1536;

  (void)in_sizes; (void)n_in; (void)out_size; (void)ws_size;

  k_zero<<<1, 256, 0, stream>>>(stats);
  // NAF: LN + conv1 (WMMA)
  k_ln_conv<true><<<NPIX / 128, 256, 0, stream>>>(x, n1_w, n1_b, 1e-6f, conv1_w,
                                                  conv1_b, w1);
  // depthwise 3x3 + instnorm stats
  k_dwconv_stats<<<BATCH * 128 * 8, 256, 0, stream>>>(w1, conv2_w, conv2_b, u,
                                                      sumA, sumQ);
  // gate + DCT attention reduction
  k_gate_dct<<<BATCH * 64 * 8, 256, 0, stream>>>(u, sumA, sumQ, in_w, in_b, tb,
                                                 ydct);
  // SE MLP
  k_se<<<1, 256, 0, stream>>>(ydct, fc1_w, fc2_w, zsc);
  // conv3 -> residual -> LN -> conv4 -> gate -> conv5 (all WMMA)
  k_mid<<<NPIX / 128, 256, 0, stream>>>(tb, zsc, x, conv3_w, conv3_b, beta,
                                        n2n_w, n2n_b, conv4_w, conv4_b, conv5_w,
                                        conv5_b, yb);
  // DFFN: LN + pin (WMMA, no bias)
  k_ln_conv<false><<<NPIX / 128, 256, 0, stream>>>(yb, ln_w, ln_b, 1e-5f, pin_w,
                                                   nullptr, w1);
  // per-patch FFT filter
  k_pfft<<<(BATCH * 128 * 28 * 28) / 256, 256, 0, stream>>>(w1, fft_w, u);
  // depthwise 3x3 + GELU gate
  k_dwgate<<<BATCH * 64 * 8, 256, 0, stream>>>(u, dw_w, tb);
  // pout (WMMA) + y -> out
  k_conv_add<<<NPIX / 128, 256, 0, stream>>>(tb, pout_w, yb, out);
}